// DetectLayer_73735998538524
// MI455X (gfx1250) — compile-verified
//
#include <hip/hip_runtime.h>
#include <hip/hip_bf16.h>

typedef __attribute__((ext_vector_type(2))) float v2f;
typedef __attribute__((ext_vector_type(8))) float v8f;

constexpr int kNB = 8, kNA = 3, kNH = 160, kNW = 160, kNC = 80, kG = 32;
constexpr int kNCell = kNA * kNH * kNW;            // 76800 cells per batch
constexpr int kBlocksPerBatch = kNCell / 256;      // 300
constexpr int kK1Blocks = kNB * kBlocksPerBatch;   // 2400
constexpr int kK2Blocks = kNB * kG;                // 256
constexpr int kNPart = kK1Blocks + kK2Blocks;      // 2656
constexpr int kNPartPad = 2688;                    // 42 * 64 (WMMA chunks)

__device__ __forceinline__ float sigmoidf(float x) { return 1.0f / (1.0f + expf(-x)); }
__device__ __forceinline__ float bcef(float x, float t) {
  return fmaxf(x, 0.0f) - x * t + log1pf(expf(-fabsf(x)));
}

// ---------------- Kernel 1: fused per-cell pass ----------------
// decode bbox, class max/argmax, score, IoU vs 32 GTs, conf BCE -> block partial.
// Invalid GTs are staged as degenerate boxes (inter==0 always), which is exactly
// equivalent to the reference's -1 sentinel + has_gt select since valid IoUs >= 0.
__global__ __launch_bounds__(256) void detect_cell_kernel(
    const float4* __restrict__ t_bbox, const float* __restrict__ conf,
    const float* __restrict__ cls, const float4* __restrict__ gt,
    const unsigned char* __restrict__ gt_mask,
    float4* __restrict__ out_pbbox, float* __restrict__ out_clsidx,
    float* __restrict__ out_score, float* __restrict__ partial) {
  __shared__ float sx1[kG], sy1[kG], sx2[kG], sy2[kG], sar[kG];
  __shared__ float red[256];

  const int tid = threadIdx.x;
  const int b = blockIdx.x / kBlocksPerBatch;
  const int n = (blockIdx.x % kBlocksPerBatch) * 256 + tid;
  const int cell = b * kNCell + n;

  if (tid < kG) {  // stage this batch's GT boxes as corners+area in LDS
    float4 gb = gt[b * kG + tid];
    bool v = gt_mask[b * kG + tid] != 0;
    sx1[tid] = v ? gb.x - gb.z * 0.5f : 1e30f;
    sx2[tid] = v ? gb.x + gb.z * 0.5f : -1e30f;
    sy1[tid] = v ? gb.y - gb.w * 0.5f : 1e30f;
    sy2[tid] = v ? gb.y + gb.w * 0.5f : -1e30f;
    sar[tid] = v ? gb.z * gb.w : 0.0f;
  }
  __syncthreads();

  const int a = n / (kNH * kNW);
  const int rem = n % (kNH * kNW);
  const int j = rem / kNW;
  const int i = rem % kNW;

  const float anchw[3] = {10.0f, 16.0f, 33.0f};
  const float anchh[3] = {13.0f, 30.0f, 23.0f};

  // prefetch this cell's 320B class row (global_prefetch_b8); consumed after
  // the IoU loop, so the fetch latency hides behind ~450 VALU ops
  const float* clsrow = cls + (size_t)cell * kNC;
  __builtin_prefetch(clsrow, 0, 1);

  // decode predicted bbox
  float4 t = t_bbox[cell];
  float sxv = sigmoidf(t.x), syv = sigmoidf(t.y);
  float swv = sigmoidf(t.z), shv = sigmoidf(t.w);
  float px = (sxv * 2.0f - 0.5f + (float)i) * 8.0f;
  float py = (syv * 2.0f - 0.5f + (float)j) * 8.0f;
  float w2 = swv * 2.0f, h2 = shv * 2.0f;
  float pw = w2 * w2 * anchw[a];
  float ph = h2 * h2 * anchh[a];
  out_pbbox[cell] = make_float4(px, py, pw, ph);

  float ax1 = px - pw * 0.5f, ax2 = px + pw * 0.5f;
  float ay1 = py - ph * 0.5f, ay2 = py + ph * 0.5f;
  float parea = pw * ph;

  // branch-free max-IoU: track best as fraction, cross-multiplied compare
  // (iou_g = inter/(union+1e-16); one division at the end)
  float best_i = 0.0f, best_u = 1.0f;
#pragma unroll 8
  for (int g = 0; g < kG; ++g) {
    float iw = fmaxf(fminf(ax2, sx2[g]) - fmaxf(ax1, sx1[g]), 0.0f);
    float ih = fmaxf(fminf(ay2, sy2[g]) - fmaxf(ay1, sy1[g]), 0.0f);
    float inter = iw * ih;
    float u = parea + sar[g] - inter + 1e-16f;
    bool upd = inter * best_u > best_i * u;
    best_i = upd ? inter : best_i;
    best_u = upd ? u : best_u;
  }
  float tconf = best_i / best_u;

  float xc = conf[cell];
  float loss_c = bcef(xc, tconf);
  float pconf = sigmoidf(xc);

  // class max/argmax (first-max tie-break; sigmoid is monotone)
  const float4* cp = (const float4*)clsrow;
  float mv = -3.402823466e+38f;
  int mi = 0;
#pragma unroll 5
  for (int k = 0; k < kNC / 4; ++k) {
    float4 v = cp[k];
    if (v.x > mv) { mv = v.x; mi = 4 * k + 0; }
    if (v.y > mv) { mv = v.y; mi = 4 * k + 1; }
    if (v.z > mv) { mv = v.z; mi = 4 * k + 2; }
    if (v.w > mv) { mv = v.w; mi = 4 * k + 3; }
  }
  out_clsidx[cell] = (float)mi;
  out_score[cell] = pconf * sigmoidf(mv);

  // deterministic block reduction of conf loss
  red[tid] = loss_c;
  __syncthreads();
  for (int s = 128; s > 0; s >>= 1) {
    if (tid < s) red[tid] += red[tid + s];
    __syncthreads();
  }
  if (tid == 0) partial[blockIdx.x] = red[0];
}

// ---------------- Kernel 2: per-GT matching + box/class BCE ----------------
__global__ __launch_bounds__(128) void detect_gt_kernel(
    const float4* __restrict__ t_bbox, const float* __restrict__ cls,
    const float4* __restrict__ gt, const int* __restrict__ gt_cls,
    const unsigned char* __restrict__ gt_mask, float* __restrict__ partial) {
  __shared__ int s_valid, s_cell;
  __shared__ float s_lbox;
  __shared__ float red[128];
  const int tid = threadIdx.x;
  const int b = blockIdx.x >> 5;
  const int g = blockIdx.x & 31;

  // block 0 zeroes the WMMA pad region [kNPart, kNPartPad)
  if (blockIdx.x == 0 && tid >= 96) partial[kK2Blocks + (tid - 96)] = 0.0f;

  if (tid == 0) {
    const float aw[9] = {10.f, 16.f, 33.f, 30.f, 62.f, 59.f, 116.f, 156.f, 373.f};
    const float ah[9] = {13.f, 30.f, 23.f, 61.f, 45.f, 119.f, 90.f, 198.f, 326.f};
    float4 gb = gt[b * kG + g];
    float gw = gb.z, gh = gb.w;
    float best = -1.0f;
    int bi = 0;
    for (int aa = 0; aa < 9; ++aa) {  // wh-IoU argmax, first-max tie-break
      float inter = fminf(gw, aw[aa]) * fminf(gh, ah[aa]);
      float uni = gw * gh + aw[aa] * ah[aa] - inter;
      float r = inter / (uni + 1e-16f);
      if (r > best) { best = r; bi = aa; }
    }
    int valid = (gt_mask[b * kG + g] != 0) && (bi < kNA);
    s_valid = valid;
    if (valid) {
      int ta = bi;  // bi < 3 so bi % 3 == bi
      int ti = (int)(gb.x * 0.125f);
      ti = ti < 0 ? 0 : (ti > kNW - 1 ? kNW - 1 : ti);
      int tj = (int)(gb.y * 0.125f);
      tj = tj < 0 ? 0 : (tj > kNH - 1 ? kNH - 1 : tj);
      int cellb = ((b * kNA + ta) * kNH + tj) * kNW + ti;
      s_cell = cellb;
      float4 tb = t_bbox[cellb];
      float tx = (fmodf(gb.x * 0.125f, 1.0f) + 0.5f) * 0.5f;
      float ty = (fmodf(gb.y * 0.125f, 1.0f) + 0.5f) * 0.5f;
      float twv = sqrtf(gb.z / aw[ta]) * 0.5f;
      float thv = sqrtf(gb.w / ah[ta]) * 0.5f;
      s_lbox = bcef(tb.x, tx) + bcef(tb.y, ty) + bcef(tb.z, twv) + bcef(tb.w, thv);
    }
  }
  __syncthreads();
  if (!s_valid) {
    if (tid == 0) partial[blockIdx.x] = 0.0f;
    return;
  }
  float v = 0.0f;
  if (tid < kNC) {
    float xl = cls[(size_t)s_cell * kNC + tid];
    float tgt = (tid == gt_cls[b * kG + g]) ? 1.0f : 0.0f;
    v = bcef(xl, tgt);
  }
  red[tid] = v;
  __syncthreads();
  for (int s = 64; s > 0; s >>= 1) {
    if (tid < s) red[tid] += red[tid + s];
    __syncthreads();
  }
  if (tid == 0) partial[blockIdx.x] = s_lbox + red[0] * (1.0f / kNC);
}

// ---------------- Kernel 3: WMMA dot-with-ones reduction ----------------
// One wave32 block. A holds 64 partials per step (16x4 f32 layout:
// lanes 0-15 -> M=lane,K={0,1}; lanes 16-31 -> M=lane-16,K={2,3}); B = ones,
// C accumulates row sums across chunks. Buffer is pre-padded with zeros to a
// multiple of 64, so every lane does one unconditional b64 load per step.
__global__ __launch_bounds__(32) void detect_reduce_kernel(
    const v2f* __restrict__ partials2, float* __restrict__ out_loss) {
  const int lane = threadIdx.x;
  const int m = lane & 15;
  const int koff = (lane < 16) ? 0 : 1;  // which K-pair (in float2 units)
  v2f bmat;
  bmat.x = 1.0f;
  bmat.y = 1.0f;
  v8f c = {};
#pragma unroll 2
  for (int base = 0; base < kNPartPad / 2; base += 32) {  // 32 float2 = 64 floats
    v2f amat = partials2[base + m * 2 + koff];
    c = __builtin_amdgcn_wmma_f32_16x16x4_f32(false, amat, false, bmat,
                                              (short)0, c, false, false);
  }
  // c[r] at lane L = row-sum for M = r + 8*(L>=16), replicated across N
  float s = c[0] + c[1] + c[2] + c[3] + c[4] + c[5] + c[6] + c[7];
  float tot = __shfl(s, 0, 32) + __shfl(s, 16, 32);
  if (lane == 0) *out_loss = tot * (1.0f / (float)kNB);
}

extern "C" void kernel_launch(void* const* d_in, const int* in_sizes, int n_in,
                              void* d_out, int out_size, void* d_ws, size_t ws_size,
                              hipStream_t stream) {
  const float4* t_bbox = (const float4*)d_in[0];          // (8,3,160,160,4) f32
  const float* conf = (const float*)d_in[1];              // (8,3,160,160,1) f32
  const float* cls = (const float*)d_in[2];               // (8,3,160,160,80) f32
  const float4* gt = (const float4*)d_in[3];              // (8,32,4) f32
  const int* gtc = (const int*)d_in[4];                   // (8,32) i32
  const unsigned char* gmask = (const unsigned char*)d_in[5];  // (8,32) bool

  float* out = (float*)d_out;
  float4* o_pb = (float4*)out;                       // 8*76800*4 floats
  float* o_ci = out + (size_t)kNB * kNCell * 4;      // 8*76800
  float* o_sc = o_ci + (size_t)kNB * kNCell;         // 8*76800
  float* o_loss = o_sc + (size_t)kNB * kNCell;       // 1
  float* part = (float*)d_ws;                        // kNPartPad floats scratch

  detect_cell_kernel<<<kK1Blocks, 256, 0, stream>>>(t_bbox, conf, cls, gt, gmask,
                                                    o_pb, o_ci, o_sc, part);
  detect_gt_kernel<<<kK2Blocks, 128, 0, stream>>>(t_bbox, cls, gt, gtc, gmask,
                                                  part + kK1Blocks);
  detect_reduce_kernel<<<1, 32, 0, stream>>>((const v2f*)part, o_loss);
}